// QuantumNLPModel_5703716569519
// MI455X (gfx1250) — compile-verified
//
#include <hip/hip_runtime.h>
#include <math.h>

typedef __attribute__((ext_vector_type(2))) float v2f;
typedef __attribute__((ext_vector_type(8))) float v8f;

#define DIM 16
#define DEPTH 2
#define DECO_KEEP 0.99f      // 1 - decoherence_rate
#define DEPOL_KEEP 0.99f     // 1 - depolarization_rate
#define DEPOL_ADD 0.000625f  // depol / dim
#define MEAS_KEEP 0.99f      // 1 - measurement_error_rate
#define MEAS_ADD 0.000625f   // meas / dim

// ---------------------------------------------------------------------------
// Phase 1: per-token noisy quantum circuit -> processed state [BS, 16] in ws.
// 4096 tokens total; completely negligible vs the GEMM, so keep it simple.
// ---------------------------------------------------------------------------
__global__ void quantum_circuit_kernel(const int* __restrict__ ids,
                                       const int* __restrict__ mask,
                                       const float* __restrict__ W_embed,
                                       const float* __restrict__ gates,
                                       float* __restrict__ state_out,
                                       int BS, int S, int B) {
    __shared__ float sg[DEPTH * DIM * DIM];
    for (int i = threadIdx.x; i < DEPTH * DIM * DIM; i += blockDim.x)
        sg[i] = gates[i];
    __syncthreads();

    int t = blockIdx.x * blockDim.x + threadIdx.x;
    if (t >= BS) return;
    int s  = t % S;
    int id = ids[t];

    float emb[DIM], st[DIM];
    float nrm = 0.f;
#pragma unroll
    for (int i = 0; i < DIM; i++) {
        float x = W_embed[(size_t)id * DIM + i];
        emb[i] = x;
        nrm += x * x;
    }
    nrm = sqrtf(nrm) + 1e-12f;
#pragma unroll
    for (int i = 0; i < DIM; i++) {
        emb[i] = emb[i] / nrm;
        st[i]  = emb[i];
    }

    for (int l = 0; l < DEPTH; l++) {
        float nv[DIM];
#pragma unroll
        for (int e = 0; e < DIM; e++) {
            float acc = 0.f;
#pragma unroll
            for (int d = 0; d < DIM; d++)
                acc += st[d] * sg[l * DIM * DIM + e * DIM + d];
            acc = acc * DECO_KEEP;                // decoherence damping
            acc = DEPOL_KEEP * acc + DEPOL_ADD;   // depolarization mix
            nv[e] = acc;
        }
        float n2 = 0.f;
#pragma unroll
        for (int e = 0; e < DIM; e++) n2 += nv[e] * nv[e];
        n2 = sqrtf(n2) + 1e-12f;
#pragma unroll
        for (int e = 0; e < DIM; e++) st[e] = nv[e] / n2;
    }
#pragma unroll
    for (int e = 0; e < DIM; e++) st[e] = MEAS_KEEP * st[e] + MEAS_ADD;

    bool apply = true;
    for (int b = 0; b < B; b++) apply = apply && (mask[b * S + s] != 0);

#pragma unroll
    for (int e = 0; e < DIM; e++)
        state_out[(size_t)t * DIM + e] = apply ? st[e] : emb[e];
}

// ---------------------------------------------------------------------------
// Phase 2: out[BS, V] = state[BS,16] @ W_out[V,16]^T + b_out, via
// V_WMMA_F32_16X16X4_F32.  One wave per 16x16 output tile; K=16 -> 4 WMMAs.
//
// 32-bit A (16x4) layout: lanes 0-15: M=lane, VGPR0=K0,VGPR1=K1;
//                         lanes 16-31: M=lane-16, VGPR0=K2,VGPR1=K3.
// B (4x16) is the dual: n = lane&15, same K split; B[k][n] = W_out[n][k],
// so per-lane both A and B are contiguous float2 loads.
// C/D layout: lane l vgpr v -> (M = v + 8*(l>=16), N = l&15); bias depends
// only on N, so we seed all 8 accumulator slots with b_out[n].
// ---------------------------------------------------------------------------
__global__ void wmma_out_proj_kernel(const float* __restrict__ state,
                                     const float* __restrict__ W_out,
                                     const float* __restrict__ b_out,
                                     float* __restrict__ out,
                                     int V) {
    const int lane  = threadIdx.x;            // 0..31, full wave, no divergence
    const int ntile = blockIdx.x * blockDim.y + threadIdx.y;
    const int mtile = blockIdx.y;
    const int m0 = mtile * 16;
    const int n0 = ntile * 16;

    const int r  = lane & 15;          // M row for A / N column for B
    const int kb = (lane >> 4) * 2;    // K sub-offset within a K=4 chunk

    const float* arow = state + (size_t)(m0 + r) * DIM + kb;
    const float* brow = W_out + (size_t)(n0 + r) * DIM + kb;

    v2f a0 = *(const v2f*)(arow + 0);
    v2f a1 = *(const v2f*)(arow + 4);
    v2f a2 = *(const v2f*)(arow + 8);
    v2f a3 = *(const v2f*)(arow + 12);
    v2f b0 = *(const v2f*)(brow + 0);
    v2f b1 = *(const v2f*)(brow + 4);
    v2f b2 = *(const v2f*)(brow + 8);
    v2f b3 = *(const v2f*)(brow + 12);

    float bias = b_out[n0 + r];
    v8f acc = {bias, bias, bias, bias, bias, bias, bias, bias};

    acc = __builtin_amdgcn_wmma_f32_16x16x4_f32(false, a0, false, b0,
                                                (short)0, acc, false, false);
    acc = __builtin_amdgcn_wmma_f32_16x16x4_f32(false, a1, false, b1,
                                                (short)0, acc, false, false);
    acc = __builtin_amdgcn_wmma_f32_16x16x4_f32(false, a2, false, b2,
                                                (short)0, acc, false, false);
    acc = __builtin_amdgcn_wmma_f32_16x16x4_f32(false, a3, false, b3,
                                                (short)0, acc, false, false);

    const int mbase = m0 + ((lane >> 4) << 3);
    float* op = out + (size_t)mbase * V + n0 + r;
#pragma unroll
    for (int v = 0; v < 8; v++)
        op[(size_t)v * V] = acc[v];
}

// ---------------------------------------------------------------------------
extern "C" void kernel_launch(void* const* d_in, const int* in_sizes, int n_in,
                              void* d_out, int out_size, void* d_ws, size_t ws_size,
                              hipStream_t stream) {
    const int*   ids     = (const int*)d_in[0];
    const int*   mask    = (const int*)d_in[1];
    const float* W_embed = (const float*)d_in[2];
    const float* gates   = (const float*)d_in[3];
    const float* W_out   = (const float*)d_in[4];
    const float* b_out   = (const float*)d_in[5];
    float*       out     = (float*)d_out;

    const int BS = in_sizes[0];        // B*S = 4096
    const int V  = in_sizes[5];        // 32000
    const int B  = 2;                  // per setup_inputs
    const int S  = BS / B;

    float* state = (float*)d_ws;       // BS*16 floats = 256 KB

    // Phase 1: quantum circuit (trivial cost)
    quantum_circuit_kernel<<<(BS + 255) / 256, 256, 0, stream>>>(
        ids, mask, W_embed, gates, state, BS, S, B);

    // Phase 2: WMMA GEMM, one wave per 16x16 tile.
    // 4 waves per block: grid = (V/64, BS/16) = (500, 256).
    dim3 block(32, 4, 1);
    dim3 grid(V / (16 * 4), BS / 16, 1);
    wmma_out_proj_kernel<<<grid, block, 0, stream>>>(state, W_out, b_out, out, V);
}